// Encoder_15298673509055
// MI455X (gfx1250) — compile-verified
//
#include <hip/hip_runtime.h>

typedef __attribute__((ext_vector_type(2))) float v2f;
typedef __attribute__((ext_vector_type(8))) float v8f;

#define LN_EPS 1e-5f

// ---------- WMMA helpers (V_WMMA_F32_16X16X4_F32, wave32) ----------
__device__ __forceinline__ v8f wmma4(v2f a, v2f b, v8f c) {
  // 8 args: (neg_a, A, neg_b, B, c_mod, C, reuse_a, reuse_b)
  return __builtin_amdgcn_wmma_f32_16x16x4_f32(false, a, false, b, (short)0, c,
                                               false, false);
}

// A 16x4 layout: lane = M (0..15, replicated to 16..31), K = vgpr + 2*(lane>=16)
__device__ __forceinline__ v2f ldA(const float* m, int ld, int r0, int c0,
                                   int l16, int h) {
  v2f a;
  a.x = m[(r0 + l16) * ld + c0 + 2 * h + 0];
  a.y = m[(r0 + l16) * ld + c0 + 2 * h + 1];
  return a;
}
// B 4x16 layout (K x N): N = lane&15, K = vgpr + 2*(lane>=16); element (c,n)=m[(c0+c)*ld+n0+n]
__device__ __forceinline__ v2f ldB(const float* m, int ld, int c0, int n0,
                                   int l16, int h) {
  v2f b;
  b.x = m[(c0 + 2 * h + 0) * ld + n0 + l16];
  b.y = m[(c0 + 2 * h + 1) * ld + n0 + l16];
  return b;
}
// B from transposed source: element (c,n) = m[(n0+n)*ld + c0+c]   (for S = Q * K^T)
__device__ __forceinline__ v2f ldBT(const float* m, int ld, int c0, int n0,
                                    int l16, int h) {
  v2f b;
  b.x = m[(n0 + l16) * ld + c0 + 2 * h + 0];
  b.y = m[(n0 + l16) * ld + c0 + 2 * h + 1];
  return b;
}

// ---------- monotonic float<->uint mapping for atomicMax on floats ----------
__device__ __forceinline__ unsigned int f2key(float f) {
  unsigned int u = __float_as_uint(f);
  return (u & 0x80000000u) ? ~u : (u | 0x80000000u);
}
__device__ __forceinline__ float key2f(unsigned int k) {
  unsigned int u = (k & 0x80000000u) ? (k & 0x7FFFFFFFu) : ~k;
  return __uint_as_float(u);
}

// ---------- block reduction over 128 threads ----------
__device__ __forceinline__ float bsum(float v, float* red, int tid) {
  __syncthreads();
  red[tid] = v;
  __syncthreads();
#pragma unroll
  for (int s = 64; s > 0; s >>= 1) {
    if (tid < s) red[tid] += red[tid + s];
    __syncthreads();
  }
  return red[0];
}

// group-local (128-thread) reduction inside a 256-thread block.
// Both groups execute the same barrier sequence (uniform control flow).
__device__ __forceinline__ float bsum_g(float v, float* red, int tid, int t,
                                        int gbase) {
  __syncthreads();
  red[tid] = v;
  __syncthreads();
#pragma unroll
  for (int s = 64; s > 0; s >>= 1) {
    if (t < s) red[tid] += red[tid + s];
    __syncthreads();
  }
  return red[gbase];
}

// ---------- init: zero softmax stats ----------
__global__ void enc_init(unsigned int* gmax, float* gsum) {
  int i = blockIdx.x * blockDim.x + threadIdx.x;
  if (i < 4096) {
    gmax[i] = 0u;  // key 0 == smallest possible
    gsum[i] = 0.0f;
  }
}

// ---------- pass 1: global per-(q,k) max of S/3 over batch ----------
__global__ __launch_bounds__(128) void enc_pass1_max(
    const float* __restrict__ x, const float* __restrict__ Wq,
    const float* __restrict__ Wk, unsigned int* __restrict__ gmax, int nb) {
  __shared__ __align__(16) float sX[1024];
  __shared__ float sQ[1024], sK[1024];
  __shared__ float sWq[256], sWk[256];
  int tid = threadIdx.x;
  int w = tid >> 5, lane = tid & 31, l16 = lane & 15, h = lane >> 4;
  int r0 = 16 * w;
  for (int i = tid; i < 256; i += 128) {
    sWq[i] = Wq[i];
    sWk[i] = Wk[i];
  }
  float vmax[4][8];
#pragma unroll
  for (int kt = 0; kt < 4; kt++)
#pragma unroll
    for (int r = 0; r < 8; r++) vmax[kt][r] = -3.4e38f;
  __syncthreads();

  for (int it = 0; it < nb; ++it) {
    size_t b = (size_t)blockIdx.x * nb + it;
    const float4* xb4 = (const float4*)(x + b * 1024);
    ((float4*)sX)[tid] = xb4[tid];
    ((float4*)sX)[tid + 128] = xb4[tid + 128];
    __syncthreads();
    v2f a[4];
#pragma unroll
    for (int kk = 0; kk < 4; kk++) a[kk] = ldA(sX, 16, r0, 4 * kk, l16, h);
    v8f q = {};
    v8f k = {};
#pragma unroll
    for (int kk = 0; kk < 4; kk++) {
      q = wmma4(a[kk], ldB(sWq, 16, 4 * kk, 0, l16, h), q);
      k = wmma4(a[kk], ldB(sWk, 16, 4 * kk, 0, l16, h), k);
    }
#pragma unroll
    for (int r = 0; r < 8; r++) {
      sQ[(r0 + r + 8 * h) * 16 + l16] = q[r];
      sK[(r0 + r + 8 * h) * 16 + l16] = k[r];
    }
    __syncthreads();
    v2f qa[4];
#pragma unroll
    for (int kk = 0; kk < 4; kk++) qa[kk] = ldA(sQ, 16, r0, 4 * kk, l16, h);
#pragma unroll
    for (int kt = 0; kt < 4; kt++) {
      v8f s = {};
#pragma unroll
      for (int kk = 0; kk < 4; kk++)
        s = wmma4(qa[kk], ldBT(sK, 16, 4 * kk, 16 * kt, l16, h), s);
#pragma unroll
      for (int r = 0; r < 8; r++)
        vmax[kt][r] = fmaxf(vmax[kt][r], s[r] * (1.0f / 3.0f));
    }
    __syncthreads();
  }
#pragma unroll
  for (int kt = 0; kt < 4; kt++)
#pragma unroll
    for (int r = 0; r < 8; r++) {
      int qrow = r0 + r + 8 * h, kcol = 16 * kt + l16;
      atomicMax(&gmax[qrow * 64 + kcol], f2key(vmax[kt][r]));
    }
}

// ---------- pass 2: global per-(q,k) sum of exp(S/3 - max) over batch ----------
__global__ __launch_bounds__(128) void enc_pass2_sum(
    const float* __restrict__ x, const float* __restrict__ Wq,
    const float* __restrict__ Wk, const unsigned int* __restrict__ gmax,
    float* __restrict__ gsum, int nb) {
  __shared__ __align__(16) float sX[1024];
  __shared__ float sQ[1024], sK[1024];
  __shared__ float sWq[256], sWk[256];
  int tid = threadIdx.x;
  int w = tid >> 5, lane = tid & 31, l16 = lane & 15, h = lane >> 4;
  int r0 = 16 * w;
  for (int i = tid; i < 256; i += 128) {
    sWq[i] = Wq[i];
    sWk[i] = Wk[i];
  }
  float gm[4][8], vsum[4][8];
#pragma unroll
  for (int kt = 0; kt < 4; kt++)
#pragma unroll
    for (int r = 0; r < 8; r++) {
      int qrow = r0 + r + 8 * h, kcol = 16 * kt + l16;
      gm[kt][r] = key2f(gmax[qrow * 64 + kcol]);
      vsum[kt][r] = 0.0f;
    }
  __syncthreads();

  for (int it = 0; it < nb; ++it) {
    size_t b = (size_t)blockIdx.x * nb + it;
    const float4* xb4 = (const float4*)(x + b * 1024);
    ((float4*)sX)[tid] = xb4[tid];
    ((float4*)sX)[tid + 128] = xb4[tid + 128];
    __syncthreads();
    v2f a[4];
#pragma unroll
    for (int kk = 0; kk < 4; kk++) a[kk] = ldA(sX, 16, r0, 4 * kk, l16, h);
    v8f q = {};
    v8f k = {};
#pragma unroll
    for (int kk = 0; kk < 4; kk++) {
      q = wmma4(a[kk], ldB(sWq, 16, 4 * kk, 0, l16, h), q);
      k = wmma4(a[kk], ldB(sWk, 16, 4 * kk, 0, l16, h), k);
    }
#pragma unroll
    for (int r = 0; r < 8; r++) {
      sQ[(r0 + r + 8 * h) * 16 + l16] = q[r];
      sK[(r0 + r + 8 * h) * 16 + l16] = k[r];
    }
    __syncthreads();
    v2f qa[4];
#pragma unroll
    for (int kk = 0; kk < 4; kk++) qa[kk] = ldA(sQ, 16, r0, 4 * kk, l16, h);
#pragma unroll
    for (int kt = 0; kt < 4; kt++) {
      v8f s = {};
#pragma unroll
      for (int kk = 0; kk < 4; kk++)
        s = wmma4(qa[kk], ldBT(sK, 16, 4 * kk, 16 * kt, l16, h), s);
#pragma unroll
      for (int r = 0; r < 8; r++)
        vsum[kt][r] += __expf(s[r] * (1.0f / 3.0f) - gm[kt][r]);
    }
    __syncthreads();
  }
#pragma unroll
  for (int kt = 0; kt < 4; kt++)
#pragma unroll
    for (int r = 0; r < 8; r++) {
      int qrow = r0 + r + 8 * h, kcol = 16 * kt + l16;
      atomicAdd(&gsum[qrow * 64 + kcol], vsum[kt][r]);
    }
}

// ---------- pass 3: attention + LN + per-patch MLPs + LN ----------
// 256 threads = two independent 4-wave batch groups sharing the staged MLP
// weights; doubles waves/SIMD vs the 128-thread version (LDS-limited to one
// workgroup per WGP either way).
#define WSTRIDE 260  // 256 + 4 pad: 260 % 64 banks = 4 -> conflict-free MLP reads

__global__ __launch_bounds__(256) void enc_pass3_final(
    const float* __restrict__ x, const float* __restrict__ Wq,
    const float* __restrict__ Wk, const float* __restrict__ Wv,
    const float* __restrict__ W1, const float* __restrict__ b1,
    const float* __restrict__ W2, const float* __restrict__ b2,
    const unsigned int* __restrict__ gmax, const float* __restrict__ gsum,
    float* __restrict__ out, int nb) {
  __shared__ float sW1[64 * WSTRIDE];  // ~65 KB, staged once per block
  __shared__ float sW2[64 * WSTRIDE];  // ~65 KB
  __shared__ float sB1[1024], sB2[1024];
  __shared__ float sWq[256], sWk[256], sWv[256];
  __shared__ __align__(16) float sX[2][1024];  // x, later y   (per group)
  __shared__ float sQ[2][1024];                // Q, later h   (per group)
  __shared__ float sK[2][1024], sV[2][1024];   // (per group)
  __shared__ float sS[2][4096];                // attn         (per group)
  __shared__ float red[256];

  int tid = threadIdx.x;
  int g = tid >> 7;   // batch group 0/1 (waves 0-3 vs 4-7)
  int t = tid & 127;  // index within group
  int gbase = g << 7;
  int w = t >> 5, lane = tid & 31, l16 = lane & 15, h = lane >> 4;
  int r0 = 16 * w;
  int p = t >> 1, o0 = (t & 1) * 8;

  float* mX = sX[g];
  float* mQ = sQ[g];
  float* mK = sK[g];
  float* mV = sV[g];
  float* mS = sS[g];

  for (int i = tid; i < 16384; i += 256) {
    int pp = i >> 8, rr = i & 255;
    sW1[pp * WSTRIDE + rr] = W1[i];
    sW2[pp * WSTRIDE + rr] = W2[i];
  }
  for (int i = tid; i < 1024; i += 256) {
    sB1[i] = b1[i];
    sB2[i] = b2[i];
  }
  if (tid < 256) {
    sWq[tid] = Wq[tid];
    sWk[tid] = Wk[tid];
    sWv[tid] = Wv[tid];
  }
  // per-(q,k) softmax stats for this wave's owned elements (batch-independent)
  float gm[4][8], rs[4][8];
#pragma unroll
  for (int kt = 0; kt < 4; kt++)
#pragma unroll
    for (int r = 0; r < 8; r++) {
      int qrow = r0 + r + 8 * h, kcol = 16 * kt + l16;
      gm[kt][r] = key2f(gmax[qrow * 64 + kcol]);
      rs[kt][r] = 1.0f / gsum[qrow * 64 + kcol];
    }
  __syncthreads();

  for (int it = 0; it < nb; ++it) {
    // each group walks its own contiguous batch range
    size_t b = (size_t)blockIdx.x * (2 * nb) + (size_t)g * nb + it;
    const float4* xb4 = (const float4*)(x + b * 1024);
    ((float4*)mX)[t] = xb4[t];
    ((float4*)mX)[t + 128] = xb4[t + 128];
    __syncthreads();

    // ---- Q,K,V stripes via WMMA ----
    v2f a[4];
#pragma unroll
    for (int kk = 0; kk < 4; kk++) a[kk] = ldA(mX, 16, r0, 4 * kk, l16, h);
    v8f q = {};
    v8f k = {};
    v8f v = {};
#pragma unroll
    for (int kk = 0; kk < 4; kk++) {
      q = wmma4(a[kk], ldB(sWq, 16, 4 * kk, 0, l16, h), q);
      k = wmma4(a[kk], ldB(sWk, 16, 4 * kk, 0, l16, h), k);
      v = wmma4(a[kk], ldB(sWv, 16, 4 * kk, 0, l16, h), v);
    }
#pragma unroll
    for (int r = 0; r < 8; r++) {
      int row = r0 + r + 8 * h;
      mQ[row * 16 + l16] = q[r];
      mK[row * 16 + l16] = k[r];
      mV[row * 16 + l16] = v[r];
    }
    __syncthreads();

    // ---- S = Q K^T / 3, attn = exp(S-max)/sum ----
    v2f qa[4];
#pragma unroll
    for (int kk = 0; kk < 4; kk++) qa[kk] = ldA(mQ, 16, r0, 4 * kk, l16, h);
#pragma unroll
    for (int kt = 0; kt < 4; kt++) {
      v8f s = {};
#pragma unroll
      for (int kk = 0; kk < 4; kk++)
        s = wmma4(qa[kk], ldBT(mK, 16, 4 * kk, 16 * kt, l16, h), s);
#pragma unroll
      for (int r = 0; r < 8; r++) {
        int qrow = r0 + r + 8 * h, kcol = 16 * kt + l16;
        mS[qrow * 64 + kcol] =
            __expf(s[r] * (1.0f / 3.0f) - gm[kt][r]) * rs[kt][r];
      }
    }
    // attn rows of this stripe are produced & consumed by the same wave; the
    // DS unit keeps per-wave LDS ops in order. mV was synced above.

    // ---- z = attn @ V (contraction over 64) ----
    v8f z = {};
#pragma unroll
    for (int c = 0; c < 16; c++)
      z = wmma4(ldA(mS, 64, r0, 4 * c, l16, h), ldB(mV, 16, 4 * c, 0, l16, h),
                z);

    // ---- residual + LayerNorm over all 1024 elements (group-local) ----
    float part = 0.0f, part2 = 0.0f, tv[8];
#pragma unroll
    for (int r = 0; r < 8; r++) {
      int row = r0 + r + 8 * h;
      float tt = mX[row * 16 + l16] + z[r];
      tv[r] = tt;
      part += tt;
      part2 += tt * tt;
    }
    float s1 = bsum_g(part, red, tid, t, gbase);
    float s2 = bsum_g(part2, red, tid, t, gbase);
    float mean = s1 * (1.0f / 1024.0f);
    float rstd = rsqrtf(s2 * (1.0f / 1024.0f) - mean * mean + LN_EPS);
#pragma unroll
    for (int r = 0; r < 8; r++) {
      int row = r0 + r + 8 * h;
      mX[row * 16 + l16] = (tv[r] - mean) * rstd;  // y
    }
    __syncthreads();

    // ---- MLP layer 1 (per-patch weights, padded stride) ----
    float acc[8];
#pragma unroll
    for (int j = 0; j < 8; j++) acc[j] = sB1[p * 16 + o0 + j];
#pragma unroll
    for (int i = 0; i < 16; i++) {
      float yv = mX[p * 16 + i];
      const float* wr = &sW1[p * WSTRIDE + i * 16 + o0];
#pragma unroll
      for (int j = 0; j < 8; j++) acc[j] = fmaf(yv, wr[j], acc[j]);
    }
#pragma unroll
    for (int j = 0; j < 8; j++) mQ[p * 16 + o0 + j] = fmaxf(acc[j], 0.0f);
    __syncthreads();

    // ---- MLP layer 2 + residual ----
    float acc2[8];
#pragma unroll
    for (int j = 0; j < 8; j++) acc2[j] = sB2[p * 16 + o0 + j];
#pragma unroll
    for (int i = 0; i < 16; i++) {
      float hv = mQ[p * 16 + i];
      const float* wr = &sW2[p * WSTRIDE + i * 16 + o0];
#pragma unroll
      for (int j = 0; j < 8; j++) acc2[j] = fmaf(hv, wr[j], acc2[j]);
    }
    float t2[8];
    float q1 = 0.0f, q2 = 0.0f;
#pragma unroll
    for (int j = 0; j < 8; j++) {
      float tt = fmaxf(acc2[j], 0.0f) + mX[p * 16 + o0 + j];
      t2[j] = tt;
      q1 += tt;
      q2 += tt * tt;
    }
    // ---- final LayerNorm + store ----
    float u1 = bsum_g(q1, red, tid, t, gbase);
    float u2 = bsum_g(q2, red, tid, t, gbase);
    float m2 = u1 * (1.0f / 1024.0f);
    float r2 = rsqrtf(u2 * (1.0f / 1024.0f) - m2 * m2 + LN_EPS);
    float* op = out + b * 1024 + p * 16 + o0;
#pragma unroll
    for (int j = 0; j < 8; j++) op[j] = (t2[j] - m2) * r2;
    __syncthreads();
  }
}

extern "C" void kernel_launch(void* const* d_in, const int* in_sizes, int n_in,
                              void* d_out, int out_size, void* d_ws,
                              size_t ws_size, hipStream_t stream) {
  (void)n_in;
  (void)out_size;
  (void)ws_size;
  const float* x = (const float*)d_in[0];
  const float* Wq = (const float*)d_in[1];
  const float* Wk = (const float*)d_in[2];
  const float* Wv = (const float*)d_in[3];
  const float* W1 = (const float*)d_in[4];
  const float* b1 = (const float*)d_in[5];
  const float* W2 = (const float*)d_in[6];
  const float* b2 = (const float*)d_in[7];
  float* out = (float*)d_out;

  unsigned int* gmax = (unsigned int*)d_ws;
  float* gsum = (float*)((char*)d_ws + 4096 * sizeof(unsigned int));

  const int B = in_sizes[0] / 1024;  // 32768 batches of 64x16
  const int NB = 64;                 // batches per block (passes 1/2), and
                                     // iterations per group (pass 3)
  const int blocks12 = B / NB;       // 512
  const int blocks3 = B / (2 * NB);  // 256 (two batch groups per block)

  enc_init<<<16, 256, 0, stream>>>(gmax, gsum);
  enc_pass1_max<<<blocks12, 128, 0, stream>>>(x, Wq, Wk, gmax, NB);
  enc_pass2_sum<<<blocks12, 128, 0, stream>>>(x, Wq, Wk, gmax, gsum, NB);
  enc_pass3_final<<<blocks3, 256, 0, stream>>>(x, Wq, Wk, Wv, W1, b1, W2, b2,
                                               gmax, gsum, out, NB);
}